// MatVecProdPytorch_67757404061738
// MI455X (gfx1250) — compile-verified
//
#include <hip/hip_runtime.h>

#define M_DIM 2048
#define N_DIM 2048
#define BATCH 64

typedef float v4f __attribute__((ext_vector_type(4)));
typedef int   v4i __attribute__((ext_vector_type(4)));

#if defined(__gfx1250__) && __has_builtin(__builtin_amdgcn_global_load_async_to_lds_b128)
#define USE_ASYNC 1
typedef __attribute__((address_space(1))) v4i* gv4i_ptr;  // global int4*
typedef __attribute__((address_space(3))) v4i* lv4i_ptr;  // LDS int4*
#else
#define USE_ASYNC 0
#endif

#if __has_builtin(__builtin_amdgcn_s_wait_asynccnt)
#define WAIT_ASYNCCNT(n) __builtin_amdgcn_s_wait_asynccnt(n)
#else
#define WAIT_ASYNCCNT(n) asm volatile("s_wait_asynccnt %0" ::"n"(n) : "memory")
#endif

constexpr int NWAVES         = 8;                          // 256 threads
constexpr int ROWS_PER_WAVE  = N_DIM / NWAVES;             // 256 n-rows per wave
constexpr int ROWS_PER_STAGE = 8;                          // 8 rows = 2 KB per stage
constexpr int STAGE_FLOATS   = ROWS_PER_STAGE * BATCH;     // 512 floats
constexpr int DEPTH          = 4;                          // ring buffers per wave
constexpr int NST            = ROWS_PER_WAVE / ROWS_PER_STAGE;  // 32 stages

// out[m,b] = sum_n M[m,n,b] * V[n,b]
// M: [2048, 2048, 64] f32 (b contiguous), V: [2048, 64] f32, out: [2048, 64] f32
__global__ __launch_bounds__(256) void batched_matvec_kernel(
    const float* __restrict__ Mp, const float* __restrict__ Vp,
    float* __restrict__ out) {
  const int tid  = threadIdx.x;
  const int w    = tid >> 5;   // wave 0..7 (wave32)
  const int lane = tid & 31;
  const int b4   = lane & 15;  // which float4 of the 64-wide batch dim
  const int half = lane >> 4;  // splits the 8 rows of a stage between lane halves
  const int m    = blockIdx.x;

  const float* Mrow  = Mp + (size_t)m * (N_DIM * BATCH);
  const v4f*   Vv    = (const v4f*)Vp;
  const int    nbase = w * ROWS_PER_WAVE;

  v4f acc = (v4f)(0.0f);

#if USE_ASYNC
  // Wave-private ring: 8 waves x 4 buffers x 2 KB = 64 KB. No cross-wave
  // LDS sharing -> the per-wave ASYNCcnt is the only sync in the hot loop.
  __shared__ __align__(16) float tile[NWAVES][DEPTH][STAGE_FLOATS];

  const char* wsrc = (const char*)(Mrow + (size_t)nbase * BATCH) + lane * 16;
  float*      wdst = &tile[w][0][0] + lane * 4;

  // One stage = 8 contiguous n-rows = 2048 B: 4 async b128 ops x 32 lanes x 16 B.
  auto issue = [&](int s) {
    const char* src = wsrc + (size_t)s * (STAGE_FLOATS * 4);
    float*      dst = wdst + (s & (DEPTH - 1)) * STAGE_FLOATS;
#pragma unroll
    for (int k = 0; k < 4; ++k) {
      __builtin_amdgcn_global_load_async_to_lds_b128(
          (gv4i_ptr)(src + k * 512),
          (lv4i_ptr)(dst + k * 128),
          /*offset=*/0, /*cpol=*/0);
    }
  };

  issue(0);
  issue(1);
  issue(2);
  for (int s = 0; s < NST; ++s) {
    if (s + 3 < NST) {
      issue(s + 3);        // keep 16 b128 async ops (8 KB) in flight per wave
      WAIT_ASYNCCNT(12);   // in-order completion -> stage s's 4 ops are done
    } else {
      WAIT_ASYNCCNT(0);    // tail drain
    }
    const float* buf = &tile[w][s & (DEPTH - 1)][0];
    const int    n0  = nbase + s * ROWS_PER_STAGE;
#pragma unroll
    for (int j = 0; j < 4; ++j) {
      const int r  = half * 4 + j;                      // row 0..7 within stage
      v4f mv = *(const v4f*)(buf + r * BATCH + b4 * 4); // ds_load_b128
      v4f vv = Vv[(size_t)(n0 + r) * 16 + b4];          // V is L2-resident
      acc += mv * vv;
    }
  }
#else
  // Fallback: non-temporal 128-bit register streaming, same decomposition.
  for (int s = 0; s < NST; ++s) {
    const int n0 = nbase + s * ROWS_PER_STAGE;
#pragma unroll
    for (int j = 0; j < 4; ++j) {
      const int n = n0 + half * 4 + j;
      v4f mv = __builtin_nontemporal_load((const v4f*)(Mrow + (size_t)n * BATCH) + b4);
      v4f vv = Vv[(size_t)n * 16 + b4];
      acc += mv * vv;
    }
  }
#endif

  // Combine the two lane-halves (same b4, disjoint rows) within the wave.
  v4f o;
  o.x = __shfl_xor(acc.x, 16, 32);
  o.y = __shfl_xor(acc.y, 16, 32);
  o.z = __shfl_xor(acc.z, 16, 32);
  o.w = __shfl_xor(acc.w, 16, 32);
  acc += o;

  // Reduce the 8 disjoint n-chunks across waves (single barrier).
  __shared__ __align__(16) v4f red[NWAVES * 16];
  if (lane < 16) red[w * 16 + b4] = acc;
  __syncthreads();
  if (tid < 16) {
    v4f sum = red[tid];
#pragma unroll
    for (int g = 1; g < NWAVES; ++g) sum += red[g * 16 + tid];
    ((v4f*)out)[(size_t)m * 16 + tid] = sum;
  }
}

extern "C" void kernel_launch(void* const* d_in, const int* in_sizes, int n_in,
                              void* d_out, int out_size, void* d_ws, size_t ws_size,
                              hipStream_t stream) {
  const float* M = (const float*)d_in[0];
  const float* V = (const float*)d_in[1];
  float* out = (float*)d_out;
  (void)in_sizes; (void)n_in; (void)out_size; (void)d_ws; (void)ws_size;
  hipLaunchKernelGGL(batched_matvec_kernel, dim3(M_DIM), dim3(256), 0, stream,
                     M, V, out);
}